// LocationSensitiveAttention_47356309406407
// MI455X (gfx1250) — compile-verified
//
#include <hip/hip_runtime.h>
#include <math.h>

#define NB   64
#define NT   2048
#define AN   256     // ANNOT_DIM (K of main GEMM)
#define QD   1024
#define AD   256     // ATTN_DIM (N)
#define NF   32
#define KW   31
#define PADW 15
#define WST  292     // LDS row stride (288 + 4 pad) -> bank-conflict-free

typedef float v2f __attribute__((ext_vector_type(2)));
typedef float v8f __attribute__((ext_vector_type(8)));

// ---------------- conv: lc[b,f,t] = sum_{c,k} conv_w[f,c,k] * loc[b,c,t+k-15] --------
__global__ void lsa_conv(const float* __restrict__ loc, const float* __restrict__ conv_w,
                         float* __restrict__ lc) {
  int idx = blockIdx.x * blockDim.x + threadIdx.x;   // b*NF*NT threads
  int t = idx & (NT - 1);
  int f = (idx >> 11) & (NF - 1);
  int b = idx >> 16;
  const float* lp0 = loc + (size_t)b * 2 * NT;
  float s = 0.f;
#pragma unroll
  for (int c = 0; c < 2; ++c) {
    const float* w  = conv_w + (f * 2 + c) * KW;
    const float* lp = lp0 + c * NT;
    for (int k = 0; k < KW; ++k) {
      int tt = t + k - PADW;
      if (tt >= 0 && tt < NT) s += w[k] * lp[tt];
    }
  }
  lc[idx] = s;
}

// ------- pqb[b,a] = query[b,:]·query_w[a,:] + query_b[a] + annot_b[a] + loc_b[a] -----
__global__ void lsa_pq(const float* __restrict__ query, const float* __restrict__ query_w,
                       const float* __restrict__ query_b, const float* __restrict__ annot_b,
                       const float* __restrict__ loc_b, float* __restrict__ pqb) {
  int a = threadIdx.x;          // 256 threads
  int b = blockIdx.x;           // 64 blocks
  const float* q = query + (size_t)b * QD;
  const float* w = query_w + (size_t)a * QD;
  float s = 0.f;
  for (int k = 0; k < QD; k += 4) {
    float4 qv = *(const float4*)(q + k);
    float4 wv = *(const float4*)(w + k);
    s += qv.x * wv.x + qv.y * wv.y + qv.z * wv.z + qv.w * wv.w;
  }
  pqb[b * AD + a] = s + query_b[a] + annot_b[a] + loc_b[a];
}

// ---------------- main fused GEMM + tanh + v-reduction --------------------------------
// one wave per 16-row (b,t) strip; 8 waves/block; weights staged in 292KB LDS
__global__ __launch_bounds__(256, 1)
void lsa_main(const float* __restrict__ annot, const float* __restrict__ lc,
              const float* __restrict__ pqb, const float* __restrict__ annot_w,
              const float* __restrict__ loc_w, const float* __restrict__ v_w,
              float* __restrict__ out) {
  __shared__ float Wt[AD * WST];   // Wt[n][k]: k<256 = annot_w[n][k], k>=256 = loc_w[n][k-256]

  for (int i = threadIdx.x; i < AD * AN / 4; i += 256) {
    int n = i >> 6, k4 = i & 63;
    float4 v = ((const float4*)annot_w)[i];
    *(float4*)&Wt[n * WST + (k4 << 2)] = v;
  }
  for (int i = threadIdx.x; i < AD * NF / 4; i += 256) {
    int n = i >> 3, k4 = i & 7;
    float4 v = ((const float4*)loc_w)[i];
    *(float4*)&Wt[n * WST + AN + (k4 << 2)] = v;
  }
  __syncthreads();

  const int wave  = threadIdx.x >> 5;
  const int lane  = threadIdx.x & 31;
  const int strip = blockIdx.x * 8 + wave;          // 8192 strips total
  const int b     = strip >> 7;                     // 128 strips per batch
  const int t0    = (strip & 127) << 4;
  const int ln    = lane & 15;
  const int hi    = lane >> 4;                      // K-half select per ISA A/B layout

  const float* Arow = annot + ((size_t)(b * NT + t0 + ln)) * AN;  // this lane's M row
  const float* lcB  = lc + (size_t)b * NF * NT + t0 + ln;

  v8f zero = {0.f, 0.f, 0.f, 0.f, 0.f, 0.f, 0.f, 0.f};
  v8f acc[16];
#pragma unroll
  for (int nt = 0; nt < 16; ++nt) acc[nt] = zero;

  // ---- annot part: K = 0..255 ----
  for (int k0 = 0; k0 < AN; k0 += 4) {
    v2f afrag = *(const v2f*)(Arow + k0 + 2 * hi);
#pragma unroll
    for (int nt = 0; nt < 16; ++nt) {
      v2f bfrag = *(const v2f*)&Wt[(nt * 16 + ln) * WST + k0 + 2 * hi];
      acc[nt] = __builtin_amdgcn_wmma_f32_16x16x4_f32(
          false, afrag, false, bfrag, (short)0, acc[nt], false, false);
    }
  }
  // ---- location part: K = 256..287 (lc is [b][f][t], strided in f) ----
#pragma unroll
  for (int f0 = 0; f0 < NF; f0 += 4) {
    int f = f0 + 2 * hi;
    v2f afrag;
    afrag.x = lcB[(size_t)f * NT];
    afrag.y = lcB[(size_t)(f + 1) * NT];
#pragma unroll
    for (int nt = 0; nt < 16; ++nt) {
      v2f bfrag = *(const v2f*)&Wt[(nt * 16 + ln) * WST + AN + f];
      acc[nt] = __builtin_amdgcn_wmma_f32_16x16x4_f32(
          false, afrag, false, bfrag, (short)0, acc[nt], false, false);
    }
  }

  // ---- epilogue: tanh(acc + pq + bias) · v_w, reduce over n --------------------------
  float rowAcc[8];
#pragma unroll
  for (int r = 0; r < 8; ++r) rowAcc[r] = 0.f;
  const float* pb = pqb + b * AD;
#pragma unroll
  for (int nt = 0; nt < 16; ++nt) {
    int n = nt * 16 + ln;
    float add = pb[n];
    float vw  = v_w[n];
#pragma unroll
    for (int r = 0; r < 8; ++r)
      rowAcc[r] += tanhf(acc[nt][r] + add) * vw;
  }
  // reduce the 16 lanes of each half (C/D layout: lanes0-15 hold m=r, lanes16-31 m=r+8)
#pragma unroll
  for (int r = 0; r < 8; ++r) {
    float s = rowAcc[r];
    s += __shfl_xor(s, 1, 32);
    s += __shfl_xor(s, 2, 32);
    s += __shfl_xor(s, 4, 32);
    s += __shfl_xor(s, 8, 32);
    rowAcc[r] = s;
  }
  if (ln == 0) {
    float* op = out + (size_t)b * NT + t0 + hi * 8;
#pragma unroll
    for (int r = 0; r < 8; ++r) op[r] = rowAcc[r];
  }
}

extern "C" void kernel_launch(void* const* d_in, const int* in_sizes, int n_in,
                              void* d_out, int out_size, void* d_ws, size_t ws_size,
                              hipStream_t stream) {
  const float* annot   = (const float*)d_in[0];
  const float* query   = (const float*)d_in[1];
  const float* loc     = (const float*)d_in[2];
  const float* conv_w  = (const float*)d_in[3];
  const float* loc_w   = (const float*)d_in[4];
  const float* loc_b   = (const float*)d_in[5];
  const float* query_w = (const float*)d_in[6];
  const float* query_b = (const float*)d_in[7];
  const float* annot_w = (const float*)d_in[8];
  const float* annot_b = (const float*)d_in[9];
  const float* v_w     = (const float*)d_in[10];
  float* out = (float*)d_out;

  float* lc  = (float*)d_ws;                         // 64*32*2048 floats = 16 MB
  float* pqb = lc + (size_t)NB * NF * NT;            // 64*256 floats

  lsa_conv<<<(NB * NF * NT) / 256, 256, 0, stream>>>(loc, conv_w, lc);
  lsa_pq<<<NB, AD, 0, stream>>>(query, query_w, query_b, annot_b, loc_b, pqb);
  lsa_main<<<(NB * NT / 16) / 8, 256, 0, stream>>>(annot, lc, pqb, annot_w, loc_w, v_w, out);
}